// BiologicalRNN_5566277615711
// MI455X (gfx1250) — compile-verified
//
#include <hip/hip_runtime.h>
#include <math.h>

// ---------------------------------------------------------------------------
// BiologicalRNN on MI455X (gfx1250, wave32, WMMA + TDM, double-buffered LDS)
//   h_{t+1} = h + alpha*(-h + x_t@W_hi_eff^T + elu(h)@W_hh^T + b)
//   out_t   = h_{t+1} @ W_oh_eff^T    (only columns j>=H/2 contribute)
// Recurrent GEMM on V_WMMA_F32_16X16X4_F32 (f32 matrix pipe: no bf16 error
// compounding over the 512-step scan). W_hh (4MB) + h (1MB) are L2-resident,
// so the scan is matrix-pipe/latency bound. W_hh/W_hi/x tiles are staged via
// the Tensor Data Mover (tensor_load_to_lds + s_wait_tensorcnt); TDM pad makes
// a conflict-free LDS row stride of 36 floats and TDM OOB-zero-fill implements
// the K=66 padding and the W_hi row mask. LDS is double-buffered so the TDM
// DMA and the elu staging of chunk c+1 overlap the WMMA chain of chunk c,
// with one barrier per chunk.
// ---------------------------------------------------------------------------

typedef __attribute__((ext_vector_type(2))) float v2f;
typedef __attribute__((ext_vector_type(8))) float v8f;

#define B_SZ   256
#define T_SZ   512
#define H_SZ   1024
#define HALF   512
#define DIN    66      // IN_COLS = 32+32+2
#define OUT_SZ 8
#define ALPHA  0.2f    // DT/TAU

#define TILE_M 32      // batch rows per block
#define TILE_N 64      // hidden-out cols per block
#define TILE_K 32      // K chunk staged in LDS
#define NCHUNK (H_SZ / TILE_K)
#define LDSS   36      // LDS row stride (floats): 32 data + 4 pad
                       // gcd(36,64)=4 -> 16 distinct banks over 16 rows; 16B aligned

// ---------------- Tensor Data Mover (D# per ISA ch.8) ----------------------
#if defined(__AMDGCN__) && __has_builtin(__builtin_amdgcn_tensor_load_to_lds) && \
    __has_builtin(__builtin_amdgcn_s_wait_tensorcnt)
#define HAS_TDM 1
typedef __attribute__((ext_vector_type(4))) unsigned int u32x4;
typedef __attribute__((ext_vector_type(8))) int          i32x8;
typedef __attribute__((ext_vector_type(4))) int          i32x4;

// 2D f32 tile load Global->LDS. Rows land at LDS stride 36 floats via TDM pad
// (pad_interval = 32 dwords, pad_amount = 4 dwords). Columns/rows beyond
// tensor_d0/tensor_d1 are zero-filled by TDM OOB handling.
__device__ __forceinline__ void tdm_load_tile_2d(const float* gsrc, float* lds_dst,
                                                 unsigned tensor_d0, unsigned tensor_d1,
                                                 unsigned stride0,
                                                 unsigned tile_d0, unsigned tile_d1)
{
    unsigned long long ga  = (unsigned long long)(uintptr_t)gsrc;
    unsigned           lds = (unsigned)(uintptr_t)lds_dst;  // low 32b of flat = LDS offset
    u32x4 g0;
    g0[0] = 1u;                                     // count=1, user mode, no gather
    g0[1] = lds;                                    // lds_addr
    g0[2] = (unsigned)ga;                           // global_addr[31:0]
    g0[3] = (unsigned)((ga >> 32) & 0x01FFFFFFu)    // global_addr[56:32]
          | 0x80000000u;                            // type = 2 ("image")
    i32x8 g1;
    g1[0] = (int)((2u << 16)                        // data_size = 4 bytes
                | (1u << 20)                        // pad_enable
                | (4u << 22)                        // pad_interval: 32 dwords
                | (3u << 25));                      // pad_amount: 4 dwords
    g1[1] = (int)(tensor_d0 << 16);                 // tensor_dim0[15:0]
    g1[2] = (int)((tensor_d0 >> 16) | (tensor_d1 << 16));
    g1[3] = (int)((tensor_d1 >> 16) | (tile_d0 << 16));   // tile_dim0
    g1[4] = (int)(tile_d1 & 0xFFFFu);               // tile_dim1 (tile_dim2 = 0)
    g1[5] = (int)stride0;                           // tensor_dim0_stride[31:0]
    g1[6] = 0;
    g1[7] = 0;
    i32x4 g2 = {1, 0, 0, 0};                        // tensor_dim2=1 (unused), no iterate
    i32x4 g3 = {0, 0, 0, 0};
    i32x8 g4 = {0, 0, 0, 0, 0, 0, 0, 0};            // extra group (clang-23 6-arg form)
    __builtin_amdgcn_tensor_load_to_lds(g0, g1, g2, g3, g4, 0);
}
#else
#define HAS_TDM 0
#endif

// Branchless fast elu: v_exp_f32 + v_cndmask, no EXEC branches.
__device__ __forceinline__ float elu_f(float v) {
    float e = __expf(v) - 1.0f;
    return v > 0.0f ? v : e;
}

// ---------------------------------------------------------------------------
// One recurrence step. Grid (H/TILE_N, B/TILE_M) = (16,8); block 256 = 8 waves,
// each wave owns one 16x16 f32 WMMA tile of the 32x64 block tile.
__global__ __launch_bounds__(256)
void rnn_step_kernel(const float* __restrict__ h_cur,
                     float* __restrict__       h_nxt,
                     const float* __restrict__ W_hh,   // [H][H]
                     const float* __restrict__ W_hi,   // [H][DIN] (rows >= HALF masked)
                     const float* __restrict__ x,      // [B][T][DIN]
                     const float* __restrict__ bias,   // [H]
                     int t)
{
    __shared__ float lA[2][TILE_M * LDSS];
    __shared__ float lB[2][TILE_N * LDSS];

    const int tid  = threadIdx.x;
    const int lane = tid & 31;
    const int wid  = tid >> 5;

    const int bn0 = blockIdx.x * TILE_N;   // hidden-output tile origin (N / j)
    const int bm0 = blockIdx.y * TILE_M;   // batch tile origin (M / b)

    const int wm = (wid >> 2) << 4;        // 0 or 16
    const int wn = (wid & 3) << 4;         // 0,16,32,48
    // f32 WMMA operand layout (ISA 7.12.2): lanes 0-15 hold K{0,1}, lanes 16-31 K{2,3}
    const int colbase = (lane >> 4) << 1;
    const int mrow = wm + (lane & 15);
    const int nrow = wn + (lane & 15);

    v8f acc = {};

    // Stage chunk c of the W_hh B-tile (64 x 32) into lB[buf].
    auto stage_B = [&](int c, int buf) {
        const int k0 = c * TILE_K;
#if HAS_TDM
        if (wid == 0)
            tdm_load_tile_2d(&W_hh[(size_t)bn0 * H_SZ + k0], lB[buf],
                             (unsigned)(H_SZ - k0), (unsigned)(H_SZ - bn0),
                             H_SZ, TILE_K, TILE_N);
#else
        const int r0 = tid >> 3, c4 = (tid & 7) << 2;
        #pragma unroll
        for (int i = 0; i < 2; ++i) {
            const int r = r0 + (i << 5);
            *(float4*)&lB[buf][r * LDSS + c4] =
                *(const float4*)&W_hh[(size_t)(bn0 + r) * H_SZ + k0 + c4];
        }
#endif
    };
    // Stage chunk c of the elu(h) A-tile (32 x 32): one b128 load + 4 elu/thread.
    auto stage_A = [&](int c, int buf) {
        const int r = tid >> 3, c4 = (tid & 7) << 2;
        float4 v = *(const float4*)&h_cur[(size_t)(bm0 + r) * H_SZ + c * TILE_K + c4];
        v.x = elu_f(v.x); v.y = elu_f(v.y); v.z = elu_f(v.z); v.w = elu_f(v.w);
        *(float4*)&lA[buf][r * LDSS + c4] = v;
    };
    // 8 chained f32 WMMAs over one staged K-chunk.
    auto compute = [&](int buf) {
        #pragma unroll
        for (int k4 = 0; k4 < TILE_K / 4; ++k4) {
            const int col = (k4 << 2) + colbase;
            v2f a = *(const v2f*)&lA[buf][mrow * LDSS + col];
            v2f b = *(const v2f*)&lB[buf][nrow * LDSS + col];
            acc = __builtin_amdgcn_wmma_f32_16x16x4_f32(
                      false, a, false, b, (short)0, acc, false, false);
        }
    };

    // -------- phase 1: acc += elu(h) @ W_hh^T, double-buffered ------------
    stage_B(0, 0);
    stage_A(0, 0);
#if HAS_TDM
    if (wid == 0) __builtin_amdgcn_s_wait_tensorcnt(0);
#endif
    __syncthreads();

    for (int c = 0; c < NCHUNK; ++c) {
        const int cur = c & 1;
        if (c + 1 < NCHUNK) {           // prefetch next chunk into the other buffer
            stage_B(c + 1, cur ^ 1);
            stage_A(c + 1, cur ^ 1);
        }
        compute(cur);                   // overlaps with TDM DMA + staged stores
#if HAS_TDM
        if (wid == 0) __builtin_amdgcn_s_wait_tensorcnt(0);
#endif
        __syncthreads();                // next buffer ready AND current buffer free
    }

    // -------- phase 2: acc += x[:,t,:] @ W_hi_eff^T (only j < HALF) -------
    if (bn0 < HALF) {   // block-uniform branch: EXEC stays all-1s for WMMA
        for (int c = 0; c < 3; ++c) {          // ceil(66/32) chunks, zero-padded
            const int d0 = c * TILE_K;
#if HAS_TDM
            if (wid == 0) {
                // A: x tile [32 rows (batch) x 32 cols], row stride T*DIN;
                //    cols >= DIN-d0 zero-filled by TDM OOB handling.
                tdm_load_tile_2d(&x[((size_t)bm0 * T_SZ + t) * DIN + d0], lA[0],
                                 (unsigned)(DIN - d0), (unsigned)(B_SZ - bm0),
                                 (unsigned)(T_SZ * DIN), TILE_K, TILE_M);
                // B: W_hi tile [64 rows (j) x 32 cols], row stride DIN.
                tdm_load_tile_2d(&W_hi[(size_t)bn0 * DIN + d0], lB[0],
                                 (unsigned)(DIN - d0), (unsigned)(HALF - bn0),
                                 DIN, TILE_K, TILE_N);
                __builtin_amdgcn_s_wait_tensorcnt(0);
            }
#else
            #pragma unroll
            for (int i = 0; i < (TILE_M * TILE_K) / 256; ++i) {
                int e = tid + i * 256, r = e >> 5, cc = e & 31;
                int d = d0 + cc;
                lA[0][r * LDSS + cc] =
                    (d < DIN) ? x[((size_t)(bm0 + r) * T_SZ + t) * DIN + d] : 0.0f;
            }
            #pragma unroll
            for (int i = 0; i < (TILE_N * TILE_K) / 256; ++i) {
                int e = tid + i * 256, r = e >> 5, cc = e & 31;
                int d = d0 + cc;
                lB[0][r * LDSS + cc] = (d < DIN) ? W_hi[(bn0 + r) * DIN + d] : 0.0f;
            }
#endif
            __syncthreads();
            compute(0);
            __syncthreads();
        }
    }

    // -------- epilogue: leaky Euler update --------------------------------
    // C/D layout: VGPR r, lanes 0-15 -> M=r, N=lane; lanes 16-31 -> M=r+8, N=lane-16
    const int jcol  = bn0 + wn + (lane & 15);
    const float bj  = bias[jcol];
    const int mbase = bm0 + wm + ((lane >> 4) << 3);
    #pragma unroll
    for (int r = 0; r < 8; ++r) {
        const int bi = mbase + r;
        const float hold = h_cur[(size_t)bi * H_SZ + jcol];
        const float hnew = hold + ALPHA * (-hold + acc[r] + bj);
        h_nxt[(size_t)bi * H_SZ + jcol] = hnew;
    }
}

// Per-step readout: outs[b][t][o] = sum_{j>=HALF} h[b][j] * W_oh[o][j]
__global__ __launch_bounds__(256)
void rnn_out_kernel(const float* __restrict__ h,
                    const float* __restrict__ W_oh,   // [O][H]
                    float* __restrict__       outs,   // [B][T][O]
                    int t)
{
    const int b    = blockIdx.x;
    const int o    = threadIdx.x >> 5;    // 8 waves -> 8 outputs
    const int lane = threadIdx.x & 31;
    float s = 0.0f;
    for (int j = HALF + lane; j < H_SZ; j += 32)
        s += h[(size_t)b * H_SZ + j] * W_oh[o * H_SZ + j];
    #pragma unroll
    for (int off = 16; off > 0; off >>= 1)
        s += __shfl_down(s, off);
    if (lane == 0)
        outs[((size_t)b * T_SZ + t) * OUT_SZ + o] = s;
}

__global__ void rnn_init_kernel(float* __restrict__ h,
                                const float* __restrict__ hidden_init)
{
    const int i = blockIdx.x * blockDim.x + threadIdx.x;
    if (i < B_SZ * H_SZ) h[i] = hidden_init[i & (H_SZ - 1)];
}

extern "C" void kernel_launch(void* const* d_in, const int* in_sizes, int n_in,
                              void* d_out, int out_size, void* d_ws, size_t ws_size,
                              hipStream_t stream)
{
    const float* x     = (const float*)d_in[0];  // [B,T,DIN]
    const float* W_hi  = (const float*)d_in[1];  // [H,DIN]
    const float* W_hh  = (const float*)d_in[2];  // [H,H]
    const float* bias  = (const float*)d_in[3];  // [H]
    const float* W_oh  = (const float*)d_in[4];  // [O,H]
    const float* hid0  = (const float*)d_in[5];  // [H]

    float* outs   = (float*)d_out;                                // [B,T,O]
    float* h_last = (float*)d_out + (size_t)B_SZ * T_SZ * OUT_SZ; // [B,H]

    float* h0 = (float*)d_ws;                    // ping
    float* h1 = h0 + (size_t)B_SZ * H_SZ;        // pong  (2 MB total)

    rnn_init_kernel<<<(B_SZ * H_SZ + 255) / 256, 256, 0, stream>>>(h0, hid0);

    dim3 grid(H_SZ / TILE_N, B_SZ / TILE_M);     // (16, 8)
    float* cur = h0;
    float* nxt = h1;
    for (int t = 0; t < T_SZ; ++t) {
        rnn_step_kernel<<<grid, 256, 0, stream>>>(cur, nxt, W_hh, W_hi, x, bias, t);
        rnn_out_kernel<<<B_SZ, 256, 0, stream>>>(nxt, W_oh, outs, t);
        float* tmp = cur; cur = nxt; nxt = tmp;
    }
    // after an even number (512) of swaps, the final state sits in `cur`
    (void)hipMemcpyAsync(h_last, cur, (size_t)B_SZ * H_SZ * sizeof(float),
                         hipMemcpyDeviceToDevice, stream);
}